// DiscUpdateUnit_55465207660625
// MI455X (gfx1250) — compile-verified
//
#include <hip/hip_runtime.h>
#include <hip/hip_bf16.h>

typedef __attribute__((ext_vector_type(2))) float v2f;
typedef __attribute__((ext_vector_type(8))) float v8f;

#define DL 128

// ---------------------------------------------------------------------------
// One wave computes a 16-row x 128-col stripe of OUT = X @ W^T with
// V_WMMA_F32_16X16X4_F32 (fp32-exact).
// K ordering trick: the reduction over K is commutative, so each 16-step
// loads a float4 at k = 8s + 4*half per lane and feeds TWO wmmas:
//   step-A frag uses (.x,.y) -> k in {8s+4h, 8s+4h+1}
//   step-B frag uses (.z,.w) -> k in {8s+4h+2, 8s+4h+3}
// A and B use the identical (half,vgpr)->k mapping, so each wmma contracts a
// consistent set of 4 k's; the union over s covers k=0..127 exactly once.
// ---------------------------------------------------------------------------
__device__ __forceinline__ void gemm_tile_acc(const float* __restrict__ X,
                                              const float* __restrict__ W,
                                              int row0, int lane, v8f acc[8]) {
  const int nl = lane & 15;
  const int half = lane >> 4;
  const float* __restrict__ xrow = X + (size_t)(row0 + nl) * DL;
  const float* __restrict__ wrow = W + (size_t)nl * DL;
  for (int s = 0; s < 16; ++s) {
    const int k = s * 8 + half * 4;
    const float4 af = *(const float4*)(xrow + k);
    float4 bf[8];
#pragma unroll
    for (int nt = 0; nt < 8; ++nt)
      bf[nt] = *(const float4*)(wrow + (size_t)(nt * 16) * DL + k);
    const v2f a0 = {af.x, af.y};
    const v2f a1 = {af.z, af.w};
#pragma unroll
    for (int nt = 0; nt < 8; ++nt) {
      const v2f b0 = {bf[nt].x, bf[nt].y};
      acc[nt] = __builtin_amdgcn_wmma_f32_16x16x4_f32(
          false, a0, false, b0, (short)0, acc[nt], false, false);
    }
#pragma unroll
    for (int nt = 0; nt < 8; ++nt) {
      const v2f b1 = {bf[nt].z, bf[nt].w};
      acc[nt] = __builtin_amdgcn_wmma_f32_16x16x4_f32(
          false, a1, false, b1, (short)0, acc[nt], false, false);
    }
  }
}

// Plain GEMM: OUT = X @ W^T.  blockDim = 32 (one wave), grid = nrows/16.
__global__ void gemm_xwt(const float* __restrict__ X, const float* __restrict__ W,
                         float* __restrict__ OUT) {
  const int lane = threadIdx.x & 31;
  const int row0 = blockIdx.x * 16;
  v8f acc[8];
#pragma unroll
  for (int nt = 0; nt < 8; ++nt) acc[nt] = (v8f){0,0,0,0,0,0,0,0};
  gemm_tile_acc(X, W, row0, lane, acc);
  const int nl = lane & 15, half = lane >> 4;
#pragma unroll
  for (int nt = 0; nt < 8; ++nt)
#pragma unroll
    for (int r = 0; r < 8; ++r)
      OUT[(size_t)(row0 + r + 8 * half) * DL + nt * 16 + nl] = acc[nt][r];
}

// Fused update: delta = relu(X@W^T + bias + msg) * (rowsum>0);
// OUT = X + delta; atomicAdd(loss_accum, sum(delta^2)).
__global__ void gemm_update(const float* __restrict__ X, const float* __restrict__ W,
                            const float* __restrict__ bias, const float* __restrict__ msg,
                            const float* __restrict__ rowsum,
                            float* __restrict__ OUT, float* __restrict__ loss_accum) {
  const int lane = threadIdx.x & 31;
  const int row0 = blockIdx.x * 16;
  v8f acc[8];
#pragma unroll
  for (int nt = 0; nt < 8; ++nt) acc[nt] = (v8f){0,0,0,0,0,0,0,0};
  gemm_tile_acc(X, W, row0, lane, acc);
  const int nl = lane & 15, half = lane >> 4;
  float lsum = 0.f;
#pragma unroll
  for (int r = 0; r < 8; ++r) {
    const int row = row0 + r + 8 * half;
    const float m = rowsum[row] > 0.f ? 1.f : 0.f;
    const size_t rb = (size_t)row * DL;
#pragma unroll
    for (int nt = 0; nt < 8; ++nt) {
      const int col = nt * 16 + nl;
      float v = acc[nt][r] + bias[col] + msg[rb + col];
      float d = v > 0.f ? v : 0.f;
      d *= m;
      lsum += d * d;
      OUT[rb + col] = X[rb + col] + d;
    }
  }
#pragma unroll
  for (int off = 16; off > 0; off >>= 1) lsum += __shfl_xor(lsum, off);
  if (lane == 0) atomicAdd(loss_accum, lsum);
}

// One wave per edge (grid-stride): msg_u[u] += v*h_i[i]; msg_i[i] += v*h_u[u].
__global__ void scatter_edges(const float* __restrict__ h_i, const float* __restrict__ h_u,
                              const int* __restrict__ u_idx, const int* __restrict__ i_idx,
                              const float* __restrict__ vals,
                              float* __restrict__ msg_u, float* __restrict__ msg_i,
                              float* __restrict__ rowsum_u, float* __restrict__ rowsum_i,
                              int n_e) {
  const int lane = threadIdx.x & 31;
  const int wave = (int)((blockIdx.x * blockDim.x + threadIdx.x) >> 5);
  const int nwaves = (int)((gridDim.x * blockDim.x) >> 5);
  for (int e = wave; e < n_e; e += nwaves) {
    const int u = u_idx[e];
    const int i = i_idx[e];
    const float v = vals[e];

    const float4 a = ((const float4*)(h_i + (size_t)i * DL))[lane];
    float* du = msg_u + (size_t)u * DL + lane * 4;
    atomicAdd(du + 0, v * a.x);
    atomicAdd(du + 1, v * a.y);
    atomicAdd(du + 2, v * a.z);
    atomicAdd(du + 3, v * a.w);

    const float4 b = ((const float4*)(h_u + (size_t)u * DL))[lane];
    float* di = msg_i + (size_t)i * DL + lane * 4;
    atomicAdd(di + 0, v * b.x);
    atomicAdd(di + 1, v * b.y);
    atomicAdd(di + 2, v * b.z);
    atomicAdd(di + 3, v * b.w);

    if (lane == 0) {
      atomicAdd(rowsum_u + u, v);
      atomicAdd(rowsum_i + i, v);
    }
  }
}

__global__ void count_mask(const float* __restrict__ rowsum, int n,
                           float* __restrict__ cnt) {
  const int idx = blockIdx.x * blockDim.x + threadIdx.x;
  float c = (idx < n && rowsum[idx] > 0.f) ? 1.f : 0.f;
#pragma unroll
  for (int off = 16; off > 0; off >>= 1) c += __shfl_xor(c, off);
  if ((threadIdx.x & 31) == 0 && c > 0.f) atomicAdd(cnt, c);
}

__global__ void finalize_loss(const float* __restrict__ s, float* __restrict__ out) {
  // s = [sum_du2, sum_di2, cnt_u, cnt_i]
  out[0] = s[0] / s[2] + s[1] / s[3];
}

extern "C" void kernel_launch(void* const* d_in, const int* in_sizes, int n_in,
                              void* d_out, int out_size, void* d_ws, size_t ws_size,
                              hipStream_t stream) {
  const float* xu   = (const float*)d_in[0];
  const float* xi   = (const float*)d_in[1];
  const int*   u_ix = (const int*)d_in[2];
  const int*   i_ix = (const int*)d_in[3];
  const float* vals = (const float*)d_in[4];
  const float* W_uu = (const float*)d_in[5];
  const float* b_uu = (const float*)d_in[6];
  const float* W_ii = (const float*)d_in[7];
  const float* b_ii = (const float*)d_in[8];
  const float* W_iu = (const float*)d_in[9];
  const float* W_ui = (const float*)d_in[10];

  const int n_u = in_sizes[0] / DL;   // 100000
  const int n_i = in_sizes[1] / DL;   // 50000
  const int n_e = in_sizes[2];        // 1600000

  float* out      = (float*)d_out;
  float* out_u    = out;                                   // xu_t_plus (temp: h_u)
  float* out_i    = out + (size_t)n_u * DL;                // xi_t_plus (temp: h_i)
  float* out_loss = out + (size_t)(n_u + n_i) * DL;        // loss_jump

  // Workspace: [su, si, cu, ci] | rowsum_u | rowsum_i | msg_u | msg_i
  float* ws        = (float*)d_ws;
  float* scal      = ws;
  float* rowsum_u  = ws + 4;
  float* rowsum_i  = rowsum_u + n_u;
  float* msg_u     = rowsum_i + n_i;
  float* msg_i     = msg_u + (size_t)n_u * DL;
  const size_t ws_floats = 4 + (size_t)n_u + (size_t)n_i + (size_t)(n_u + n_i) * DL;

  hipMemsetAsync(d_ws, 0, ws_floats * sizeof(float), stream);

  // h_i = xi @ W_iu^T ; h_u = xu @ W_ui^T  (staged in d_out regions)
  gemm_xwt<<<n_i / 16, 32, 0, stream>>>(xi, W_iu, out_i);
  gemm_xwt<<<n_u / 16, 32, 0, stream>>>(xu, W_ui, out_u);

  scatter_edges<<<4096, 256, 0, stream>>>(out_i, out_u, u_ix, i_ix, vals,
                                          msg_u, msg_i, rowsum_u, rowsum_i, n_e);

  count_mask<<<(n_u + 255) / 256, 256, 0, stream>>>(rowsum_u, n_u, scal + 2);
  count_mask<<<(n_i + 255) / 256, 256, 0, stream>>>(rowsum_i, n_i, scal + 3);

  gemm_update<<<n_u / 16, 32, 0, stream>>>(xu, W_uu, b_uu, msg_u, rowsum_u,
                                           out_u, scal + 0);
  gemm_update<<<n_i / 16, 32, 0, stream>>>(xi, W_ii, b_ii, msg_i, rowsum_i,
                                           out_i, scal + 1);

  finalize_loss<<<1, 1, 0, stream>>>(scal, out_loss);
}